// WaveletTransformLayer_6554120093997
// MI455X (gfx1250) — compile-verified
//
#include <hip/hip_runtime.h>

typedef float v2f __attribute__((ext_vector_type(2)));
typedef float v8f __attribute__((ext_vector_type(8)));

#define T_LEN     2048
#define F_N       32
#define TILE_T    128
#define ROWS_LD   140          // 128 outputs + 11-row tap reach, padded to chunk edge
#define LDSS      33           // padded LDS row stride (gcd(33,64)=1 -> conflict-free)
#define N_OUT     8165
#define OFF_D1    0
#define LEN_D1    2047
#define OFF_D2    2047
#define LEN_D2    2044
#define OFF_D3    4091
#define OFF_A3    6128
#define LEN_3     2037

// 12-tap composite box-filter coefficients (x64), computed arithmetically (no tables):
// a3 (box2*box4*box8): [1,3,5,7,8,8,8,8,7,5,3,1]
__device__ __forceinline__ float coef_a3(int j) {
    if (j < 0 || j > 11) return 0.0f;
    int v = 2 * j + 1;
    if (v > 8) v = 8;
    int w = 23 - 2 * j;
    if (w < v) v = w;
    return (float)v;
}
// d3 = ma2[i+7] - ma3[i]:  ma2 taps (x64) = [8,16,16,16,8] at j=7..11, minus a3 coeffs
__device__ __forceinline__ float coef_d3(int j) {
    if (j < 0 || j > 11) return 0.0f;
    int ma2c = 0;
    if (j == 7 || j == 11) ma2c = 8;
    else if (j > 7 && j < 11) ma2c = 16;
    return (float)ma2c - coef_a3(j);
}

__global__ __launch_bounds__(256)
void wavelet_pyramid_kernel(const float* __restrict__ x, float* __restrict__ out) {
    __shared__ float smem[ROWS_LD * LDSS];

    const int tid  = threadIdx.x;
    const int tile = blockIdx.x;       // 16 time tiles
    const int b    = blockIdx.y;       // batch
    const int t0   = tile * TILE_T;

    const float* xb = x + (size_t)b * (T_LEN * F_N);

    // ---- Stage input tile (+halo) into LDS; fully coalesced 4B/lane streams ----
    for (int idx = tid; idx < ROWS_LD * F_N; idx += 256) {
        int g = t0 * F_N + idx;                      // (t,f) contiguous in memory
        const int gmax = T_LEN * F_N - 1;
        g = (g < gmax) ? g : gmax;                   // clamp at series end (masked later)
        const int row = idx >> 5;
        const int f   = idx & 31;
        smem[row * LDSS + f] = xb[g];
    }
    __syncthreads();

    const int lane  = tid & 31;
    const int wave  = tid >> 5;
    const int nloc  = lane & 15;               // N (local time) index this lane owns
    const int khalf = (lane >> 4) << 1;        // K split: lanes 0-15 -> k0/k1, 16-31 -> k2/k3
    const int mhi   = (lane >> 4) << 3;        // D-layout M offset: 0 or 8

    // ---- Constant Toeplitz B fragments: B[k,n] = c[(4c + k) - n] / (64*2048) ----
    const float scale = 1.0f / (64.0f * 2048.0f);
    v2f B3[7], B4[7];
#pragma unroll
    for (int c = 0; c < 7; ++c) {
        const int j0 = 4 * c + khalf - nloc;
        B3[c].x = coef_d3(j0)     * scale;
        B3[c].y = coef_d3(j0 + 1) * scale;
        B4[c].x = coef_a3(j0)     * scale;
        B4[c].y = coef_a3(j0 + 1) * scale;
    }

    const int nbase = wave * 16;                       // this wave's 16-output time chunk
    float* outb = out + (size_t)b * (F_N * N_OUT);

    // ---- 12-tap filters d3/a3 via V_WMMA_F32_16X16X4_F32 (banded Toeplitz matmul) ----
    // D[m=f, n=t] = sum_p  x[t0+p, f] * c[p - n];  7 K-chunks of 4 cover the 28-wide band.
#pragma unroll
    for (int fh = 0; fh < 2; ++fh) {
        const int frow = fh * 16 + nloc;               // feature row this lane loads for A
        v8f acc3 = {};
        v8f acc4 = {};
#pragma unroll
        for (int c = 0; c < 7; ++c) {
            const int p0 = nbase + 4 * c + khalf;      // A layout: VGPR0={k0|k2}, VGPR1={k1|k3}
            v2f a;
            a.x = smem[(p0    ) * LDSS + frow];
            a.y = smem[(p0 + 1) * LDSS + frow];
            acc3 = __builtin_amdgcn_wmma_f32_16x16x4_f32(false, a, false, B3[c],
                                                         (short)0, acc3, false, false);
            acc4 = __builtin_amdgcn_wmma_f32_16x16x4_f32(false, a, false, B4[c],
                                                         (short)0, acc4, false, false);
        }
        // D layout: VGPR r -> M = r (lanes 0-15) / r+8 (lanes 16-31), N = lane&15.
        // Lanes with consecutive nloc write consecutive time addresses -> coalesced.
        const int i = t0 + nbase + nloc;
        if (i < LEN_3) {
#pragma unroll
            for (int r = 0; r < 8; ++r) {
                const int f = fh * 16 + r + mhi;
                float* ob = outb + (size_t)f * N_OUT;
                ob[OFF_D3 + i] = acc3[r];
                ob[OFF_A3 + i] = acc4[r];
            }
        }
    }

    // ---- Short filters d1 (2 taps) and d2 (5 taps) via VALU; coalesced along time ----
#pragma unroll
    for (int iter = 0; iter < 16; ++iter) {
        const int g  = iter * 256 + tid;
        const int tl = g & 127;                        // lanes sweep time -> coalesced stores
        const int f  = g >> 7;
        const int i  = t0 + tl;
        const float x0 = smem[(tl    ) * LDSS + f];
        const float x1 = smem[(tl + 1) * LDSS + f];
        if (i < LEN_D1) {
            outb[(size_t)f * N_OUT + OFF_D1 + i] = (x1 - x0) * (0.5f / 2048.0f);
        }
        if (i < LEN_D2) {
            const float x2 = smem[(tl + 2) * LDSS + f];
            const float x3 = smem[(tl + 3) * LDSS + f];
            const float x4 = smem[(tl + 4) * LDSS + f];
            outb[(size_t)f * N_OUT + OFF_D2 + i] =
                (-x0 - 2.0f * x1 - 2.0f * x2 + 2.0f * x3 + 3.0f * x4) *
                (1.0f / (8.0f * 2048.0f));
        }
    }
}

extern "C" void kernel_launch(void* const* d_in, const int* in_sizes, int n_in,
                              void* d_out, int out_size, void* d_ws, size_t ws_size,
                              hipStream_t stream) {
    (void)n_in; (void)d_ws; (void)ws_size; (void)out_size;
    const float* x = (const float*)d_in[0];
    float* out = (float*)d_out;
    const int B = in_sizes[0] / (T_LEN * F_N);         // 128
    dim3 grid(T_LEN / TILE_T, B);                      // 16 x 128 blocks
    wavelet_pyramid_kernel<<<grid, 256, 0, stream>>>(x, out);
}